// FeatureEncoder_11922829213853
// MI455X (gfx1250) — compile-verified
//
#include <hip/hip_runtime.h>
#include <math.h>

typedef float v2f __attribute__((ext_vector_type(2)));
typedef float v8f __attribute__((ext_vector_type(8)));

#define L_MAX   16
#define D_MODEL 64
#define WAVES_PER_BLOCK 8

// One wave32 per (item, variation) pair.
//   pre[d] = sum_l mask(x[l]) * W[l][d] + b[d];  out = gelu_tanh(pre)
// WMMA mapping (V_WMMA_F32_16X16X4_F32, D = A(16x4) * B(4x16) + C):
//   A[m,k] = x[4c+k]  (replicated over M rows)
//   B[k,n] = W[4c+k][16j+n]
//   => D[m,n] = out_tile_j[n]  replicated over M.
__global__ __launch_bounds__(256) void feat_enc_wmma(
    const float* __restrict__ params,   // [P, 16]
    const float* __restrict__ weights,  // [P, 16, 64] row-major (l, d)
    const float* __restrict__ bias,     // [P, 64]
    const int*   __restrict__ lengths,  // [P]
    float* __restrict__ out,            // [P, 64]
    int npairs)
{
    const int lane = threadIdx.x & 31;
    const int wave = threadIdx.x >> 5;
    const int pair = blockIdx.x * WAVES_PER_BLOCK + wave;
    if (pair >= npairs) return;          // wave-uniform: EXEC all-1s below

    const int half = lane >> 4;          // 0: lanes 0-15, 1: lanes 16-31
    const int nn   = lane & 15;          // N column within a tile
    const int bk   = half << 1;          // base K held by this half (ISA layout)

    const float* __restrict__ xp = params  + (size_t)pair * L_MAX;
    const float* __restrict__ wp = weights + (size_t)pair * (L_MAX * D_MODEL);
    const float* __restrict__ bp = bias    + (size_t)pair * D_MODEL;
    const int len = lengths[pair];

    // A operands: per K-chunk c, this lane holds x[4c+bk], x[4c+bk+1],
    // zero-masked by the ragged length (cndmask, no branch divergence).
    v2f a[4];
#pragma unroll
    for (int c = 0; c < 4; ++c) {
        const int i0 = 4 * c + bk;
        const float x0 = xp[i0];
        const float x1 = xp[i0 + 1];
        a[c][0] = (i0     < len) ? x0 : 0.0f;
        a[c][1] = (i0 + 1 < len) ? x1 : 0.0f;
    }

    // 4 N-tiles of 16 features; K=16 via 4 chained 16x16x4 WMMAs each.
    v8f acc[4];
#pragma unroll
    for (int j = 0; j < 4; ++j) {
        v8f c8 = {};
#pragma unroll
        for (int c = 0; c < 4; ++c) {
            const int r0 = 4 * c + bk;
            v2f b;
            b[0] = wp[(r0    ) * D_MODEL + 16 * j + nn];  // 64B contig / half-wave
            b[1] = wp[(r0 + 1) * D_MODEL + 16 * j + nn];
            // (neg_a, A, neg_b, B, c_mod, C, reuse_a, reuse_b)
            c8 = __builtin_amdgcn_wmma_f32_16x16x4_f32(
                false, a[c], false, b, (short)0, c8, false, false);
        }
        acc[j] = c8;
    }

    // D VGPR0: lanes 0-15 and 16-31 both hold out_tile_j[nn] (replicated over M).
    // Lane-half select stitches two tiles into 32 contiguous features.
    float pre0 = (half == 0) ? acc[0][0] : acc[1][0];   // feature = lane
    float pre1 = (half == 0) ? acc[2][0] : acc[3][0];   // feature = 32 + lane

    pre0 += bp[lane];
    pre1 += bp[32 + lane];

    // jax.nn.gelu (tanh approximation), branch-free:
    //   0.5*(1 + tanh(z)) == sigmoid(2z)
    //   gelu(x) = x * sigmoid(2*k0*(x + k1*x^3))
    // __expf -> v_mul + hardware v_exp_f32 (no divergent OCML slow path).
    const float two_k0 = 1.5957691216057308f;   // 2*sqrt(2/pi)
    const float k1     = 0.044715f;
    float z0 = two_k0 * (pre0 + k1 * pre0 * pre0 * pre0);
    float z1 = two_k0 * (pre1 + k1 * pre1 * pre1 * pre1);
    float s0 = __frcp_rn(1.0f + __expf(-z0));
    float s1 = __frcp_rn(1.0f + __expf(-z1));
    float g0 = pre0 * s0;
    float g1 = pre1 * s1;

    float* __restrict__ op = out + (size_t)pair * D_MODEL;
    op[lane]      = g0;   // 128B fully-coalesced store
    op[32 + lane] = g1;   // 128B fully-coalesced store
}

extern "C" void kernel_launch(void* const* d_in, const int* in_sizes, int n_in,
                              void* d_out, int out_size, void* d_ws, size_t ws_size,
                              hipStream_t stream) {
    const float* params  = (const float*)d_in[0];  // [N, V, 16]
    const float* weights = (const float*)d_in[1];  // [N, V, 16, 64]
    const float* bias    = (const float*)d_in[2];  // [N, V, 64]
    const int*   lengths = (const int*)  d_in[3];  // [N, V]
    float*       out     = (float*)d_out;          // [N, V, 64]

    const int npairs = in_sizes[3];                // N * V = 80000
    const int blocks = (npairs + WAVES_PER_BLOCK - 1) / WAVES_PER_BLOCK;

    feat_enc_wmma<<<blocks, WAVES_PER_BLOCK * 32, 0, stream>>>(
        params, weights, bias, lengths, out, npairs);
}